// InstanceWiseAveragePooling_42107859370186
// MI455X (gfx1250) — compile-verified
//
#include <hip/hip_runtime.h>

typedef float v2f __attribute__((ext_vector_type(2)));
typedef float v8f __attribute__((ext_vector_type(8)));

#define NBLK 1024
#define NTHR 256

// Deterministic 32-lane sum using V_WMMA_F32_16X16X4_F32 (f32 in, f32 acc).
// A layout (16x4 f32, 2 VGPRs): vgpr0 = {lanes0-15: A[m][0], lanes16-31: A[m][2]},
// vgpr1 = {A[m][1], A[m][3]}. With B = all-ones, D[m][*] = sum_k A[m][k]
// = acc[lane m] + acc[lane m+16]. Lanes 0-15 hold D rows 0-7 in their 8 D VGPRs,
// lanes 16-31 hold rows 8-15; summing D VGPRs + shfl_xor(16) yields the wave sum
// in every lane. EXEC must be all-1s at the call site (it is: no divergence here).
__device__ __forceinline__ float wave_sum_wmma(float v) {
    v2f a; a.x = v;    a.y = 0.0f;
    v2f b; b.x = 1.0f; b.y = 1.0f;
    v8f c = {};
    c = __builtin_amdgcn_wmma_f32_16x16x4_f32(
        /*neg_a=*/false, a, /*neg_b=*/false, b,
        /*c_mod=*/(short)0, c, /*reuse_a=*/false, /*reuse_b=*/false);
    float r = ((c[0] + c[1]) + (c[2] + c[3])) + ((c[4] + c[5]) + (c[6] + c[7]));
    r += __shfl_xor(r, 16, 32);   // rows 0-7 + rows 8-15
    return r;                      // full 32-lane sum, in every lane
}

// Pass 1: per-block partial class sums & counts.
// partials[b*6 + 0..2] = sum_c, partials[b*6 + 3..5] = count_c (count incl. x3 channels)
__global__ void iwap_pass1(const float* __restrict__ feats,
                           const int*   __restrict__ inst,
                           float* __restrict__ partials, int npix) {
    float accS[3] = {0.f, 0.f, 0.f};
    float accN[3] = {0.f, 0.f, 0.f};

    const int tid    = blockIdx.x * blockDim.x + threadIdx.x;
    const int stride = gridDim.x * blockDim.x;
    const int nquad  = npix >> 2;

    for (int q = tid; q < nquad; q += stride) {
        const int p = q << 2;                    // pixel index, multiple of 4
        const float4* f4 = (const float4*)(feats + (size_t)p * 3);
        float4 fa = f4[0];                       // 12 contiguous floats = 4 pixels
        float4 fb = f4[1];
        float4 fc = f4[2];
        int4 ii = *(const int4*)(inst + p);

        float s0 = fa.x + fa.y + fa.z;
        float s1 = fa.w + fb.x + fb.y;
        float s2 = fb.z + fb.w + fc.x;
        float s3 = fc.y + fc.z + fc.w;

        #pragma unroll
        for (int cc = 0; cc < 3; ++cc) {
            accS[cc] += (ii.x == cc ? s0 : 0.f) + (ii.y == cc ? s1 : 0.f)
                      + (ii.z == cc ? s2 : 0.f) + (ii.w == cc ? s3 : 0.f);
            accN[cc] += (ii.x == cc ? 3.f : 0.f) + (ii.y == cc ? 3.f : 0.f)
                      + (ii.z == cc ? 3.f : 0.f) + (ii.w == cc ? 3.f : 0.f);
        }
    }
    // tail pixels (npix % 4) — none for this shape, kept for generality
    for (int p = (nquad << 2) + tid; p < npix; p += stride) {
        const size_t fp = (size_t)p * 3;
        float s = feats[fp] + feats[fp + 1] + feats[fp + 2];
        int cl = inst[p];
        #pragma unroll
        for (int cc = 0; cc < 3; ++cc) {
            accS[cc] += (cl == cc ? s : 0.f);
            accN[cc] += (cl == cc ? 3.f : 0.f);
        }
    }

    // Wave-level reduction via f32 WMMA (all lanes active here).
    float wv[6];
    #pragma unroll
    for (int cc = 0; cc < 3; ++cc) {
        wv[cc]     = wave_sum_wmma(accS[cc]);
        wv[3 + cc] = wave_sum_wmma(accN[cc]);
    }

    // Cross-wave (8 waves / 256 threads) reduction in LDS, deterministic order.
    __shared__ float sm[8][6];
    const int lane = threadIdx.x & 31;
    const int wave = threadIdx.x >> 5;
    if (lane == 0) {
        #pragma unroll
        for (int k = 0; k < 6; ++k) sm[wave][k] = wv[k];
    }
    __syncthreads();
    if (threadIdx.x < 6) {
        float t = 0.f;
        #pragma unroll
        for (int w = 0; w < 8; ++w) t += sm[w][threadIdx.x];
        partials[(size_t)blockIdx.x * 6 + threadIdx.x] = t;
    }
}

// Pass 2: single block reduces NBLK partials (f64 accumulate) and emits 3 means.
__global__ void iwap_pass2(const float* __restrict__ partials,
                           float* __restrict__ means, int nblk) {
    __shared__ double red[6][NTHR];
    double acc[6] = {0, 0, 0, 0, 0, 0};
    for (int i = threadIdx.x; i < nblk; i += blockDim.x) {
        #pragma unroll
        for (int k = 0; k < 6; ++k) acc[k] += (double)partials[(size_t)i * 6 + k];
    }
    #pragma unroll
    for (int k = 0; k < 6; ++k) red[k][threadIdx.x] = acc[k];
    __syncthreads();
    for (int s = NTHR / 2; s > 0; s >>= 1) {
        if (threadIdx.x < (unsigned)s) {
            #pragma unroll
            for (int k = 0; k < 6; ++k) red[k][threadIdx.x] += red[k][threadIdx.x + s];
        }
        __syncthreads();
    }
    if (threadIdx.x == 0) {
        #pragma unroll
        for (int c = 0; c < 3; ++c)
            means[c] = (float)(red[c][0] / red[3 + c][0]);
    }
}

// Pass 3: out[pixel][ch] = mean[inst[pixel]] (classes partition all pixels).
__global__ void iwap_pass3(const int* __restrict__ inst,
                           const float* __restrict__ means,
                           float* __restrict__ out, int npix) {
    const float m0 = means[0], m1 = means[1], m2 = means[2];

    const int tid    = blockIdx.x * blockDim.x + threadIdx.x;
    const int stride = gridDim.x * blockDim.x;
    const int nquad  = npix >> 2;

    for (int q = tid; q < nquad; q += stride) {
        const int p = q << 2;
        int4 ii = *(const int4*)(inst + p);
        float a = (ii.x == 0) ? m0 : ((ii.x == 1) ? m1 : m2);
        float b = (ii.y == 0) ? m0 : ((ii.y == 1) ? m1 : m2);
        float c = (ii.z == 0) ? m0 : ((ii.z == 1) ? m1 : m2);
        float d = (ii.w == 0) ? m0 : ((ii.w == 1) ? m1 : m2);
        float4* o4 = (float4*)(out + (size_t)p * 3);
        o4[0] = make_float4(a, a, a, b);
        o4[1] = make_float4(b, b, c, c);
        o4[2] = make_float4(c, d, d, d);
    }
    for (int p = (nquad << 2) + tid; p < npix; p += stride) {
        int cl = inst[p];
        float m = (cl == 0) ? m0 : ((cl == 1) ? m1 : m2);
        const size_t fp = (size_t)p * 3;
        out[fp] = m; out[fp + 1] = m; out[fp + 2] = m;
    }
}

extern "C" void kernel_launch(void* const* d_in, const int* in_sizes, int n_in,
                              void* d_out, int out_size, void* d_ws, size_t ws_size,
                              hipStream_t stream) {
    const float* feats = (const float*)d_in[0];
    const int*   inst  = (const int*)d_in[1];
    float*       out   = (float*)d_out;
    const int    npix  = in_sizes[1];          // B*H*W (inst element count)

    float* partials = (float*)d_ws;            // NBLK * 6 floats
    float* means    = partials + (size_t)NBLK * 6;  // 3 floats

    iwap_pass1<<<NBLK, NTHR, 0, stream>>>(feats, inst, partials, npix);
    iwap_pass2<<<1, NTHR, 0, stream>>>(partials, means, NBLK);
    iwap_pass3<<<NBLK, NTHR, 0, stream>>>(inst, means, out, npix);
}